// Fractal2D_9414568313336
// MI455X (gfx1250) — compile-verified
//
#include <hip/hip_runtime.h>

// MI455X / gfx1250 (CDNA5, wave32) implementation.
// Stage 1: per-(image,channel,k) fractal metrics via register-bitmask flood fill
//          + LDS-atomic reductions (input fits in 192MB L2; total traffic ~0.6GB).
// Stage 2: bilinear 5x5 -> 224x224 resize expressed as Out = A * M * A^T and
//          executed with V_WMMA_F32_16X16X4_F32 (two K=4 steps per 16x16 tile).

typedef __attribute__((ext_vector_type(2))) float v2f;
typedef __attribute__((ext_vector_type(8))) float v8f;

#define IMG_B 64
#define IMG_H 224
#define IMG_W 224
#define NCH   3

static __device__ __forceinline__ int imini(int a, int b) { return a < b ? a : b; }
static __device__ __forceinline__ int imaxi(int a, int b) { return a > b ? a : b; }

// jax.image.resize bilinear weight (half-pixel centers, edge-renormalized ==
// clamped-index triangle kernel), 5 -> 224, exact in f32.
static __device__ __forceinline__ float interp_w(int i, int p) {
    float f  = (i + 0.5f) * (5.0f / 224.0f) - 0.5f;
    float lo = floorf(f);
    float t  = f - lo;
    int plo = (int)lo;
    int phi = plo + 1;
    plo = imini(imaxi(plo, 0), 4);
    phi = imini(imaxi(phi, 0), 4);
    float w = 0.0f;
    if (plo == p) w += 1.0f - t;
    if (phi == p) w += t;
    return w;
}

// ---------------- Stage 1: metrics ----------------

template <int K>
__device__ __forceinline__ void patch_stats(const float* __restrict__ in,
                                            int b, int ch, int y0, int x0,
                                            int& ones_out, int& ncomp_out, int& marea_out) {
    const float kf = (float)K;
    // center pixel (may be a zero-pad pixel)
    float center = 0.0f;
    {
        int cy = y0 + K / 2, cx = x0 + K / 2;
        if (cy >= 0 && cy < IMG_H && cx >= 0 && cx < IMG_W)
            center = in[((b * IMG_H + cy) * IMG_W + cx) * NCH + ch];
    }
    unsigned bin[K];
    int ones = 0;
#pragma unroll
    for (int r = 0; r < K; ++r) {
        unsigned m = 0;
        int y = y0 + r;
        if (y >= 0 && y < IMG_H) {
            for (int c = 0; c < K; ++c) {
                int x = x0 + c;
                float v = 0.0f;
                if (x >= 0 && x < IMG_W) v = in[((b * IMG_H + y) * IMG_W + x) * NCH + ch];
                if (fabsf(v - center) <= kf) m |= (1u << c);
            }
        } else {
            // fully padded row: every value is 0
            if (fabsf(center) <= kf) m = (1u << K) - 1u;
        }
        bin[r] = m;
        ones += __popc(m);
    }

    // 4-connected components: word-parallel flood fill on row bitmasks
    unsigned rem[K];
#pragma unroll
    for (int r = 0; r < K; ++r) rem[r] = bin[r];
    int remaining = ones, ncomp = 0, maxarea = 0;
    while (remaining > 0) {
        int r0 = -1;
        unsigned seed = 0;
#pragma unroll
        for (int r = 0; r < K; ++r) {
            if (r0 < 0 && rem[r] != 0) { r0 = r; seed = rem[r] & (0u - rem[r]); }
        }
        unsigned cur[K];
#pragma unroll
        for (int r = 0; r < K; ++r) cur[r] = (r == r0) ? seed : 0u;
        bool changed = true;
        while (changed) {
            changed = false;
#pragma unroll
            for (int r = 0; r < K; ++r) {
                unsigned g = cur[r] | (cur[r] << 1) | (cur[r] >> 1);
                if (r > 0) g |= cur[r - 1];
                if (r < K - 1) g |= cur[r + 1];
                g &= rem[r];
                if (g != cur[r]) { cur[r] = g; changed = true; }
            }
        }
        int area = 0;
#pragma unroll
        for (int r = 0; r < K; ++r) { area += __popc(cur[r]); rem[r] &= ~cur[r]; }
        remaining -= area;
        ++ncomp;
        maxarea = imaxi(maxarea, area);
    }
    ones_out  = ones;
    ncomp_out = ncomp;
    // "most common label count" includes background label 0 (k*k - ones zeros)
    marea_out = imaxi(K * K - ones, maxarea);
}

template <int K>
__device__ __forceinline__ void metrics_body(const float* __restrict__ in,
                                             float* __restrict__ mets,
                                             int kidx, int ch, int b,
                                             int* hist, int* sums) {
    const int rows = (IMG_H + K - 1) / K;
    const int P    = rows * rows;
    const int pt   = (rows * K - IMG_H) / 2;   // ph // 2
    const int kk   = K * K;

    for (int i = threadIdx.x; i < kk + 1; i += blockDim.x) hist[i] = 0;
    if (threadIdx.x == 0) { sums[0] = 0; sums[1] = 0; sums[2] = 0; }
    __syncthreads();

    int l_ncomp = 0, l_marea = 0, l_perc = 0;
    for (int p = threadIdx.x; p < P; p += blockDim.x) {
        int pr = p / rows, pc = p - pr * rows;
        int ones, ncomp, marea;
        patch_stats<K>(in, b, ch, pr * K - pt, pc * K - pt, ones, ncomp, marea);
        l_ncomp += ncomp;
        l_marea += marea;
        if ((float)ones / (float)kk >= 0.59275f) ++l_perc;
        atomicAdd(&hist[ones], 1);           // ds_add
    }
    atomicAdd(&sums[0], l_ncomp);
    atomicAdd(&sums[1], l_marea);
    atomicAdd(&sums[2], l_perc);
    __syncthreads();

    if (threadIdx.x == 0) {
        const float Pf = (float)P;
        float fd = 0.0f, m = 0.0f, m2 = 0.0f;
        for (int i = 0; i < kk; ++i) {       // bin ones==k*k dropped, per reference
            float prob = (float)hist[i] / Pf;
            float n    = (float)(i + 1);
            fd += prob / n;
            m  += prob * n;
            m2 += prob * prob * n;
        }
        float lac  = (m2 - m * m) / (m * m);
        float acn  = (float)(sums[0] / P);   // integer floor-div, per reference
        float acma = (float)(sums[1] / P);
        float acp  = (float)(sums[2] / P);
        // metric-major layout (matches _rearrange): mets[b][metric][kidx][ch]
        float* o = mets + ((size_t)b * 25) * NCH + kidx * NCH + ch;
        o[0 * 5 * NCH] = acn;
        o[1 * 5 * NCH] = acp;
        o[2 * 5 * NCH] = acma;
        o[3 * 5 * NCH] = lac;
        o[4 * 5 * NCH] = fd;
    }
}

__global__ __launch_bounds__(256)
void fractal_metrics_kernel(const float* __restrict__ in, float* __restrict__ mets) {
    __shared__ int hist[128];
    __shared__ int sums[3];
    const int kidx = blockIdx.x;   // 0..4  -> k = 3,5,7,9,11
    const int ch   = blockIdx.y;   // 0..2
    const int b    = blockIdx.z;   // 0..63
    switch (kidx) {
        case 0: metrics_body<3>(in, mets, kidx, ch, b, hist, sums); break;
        case 1: metrics_body<5>(in, mets, kidx, ch, b, hist, sums); break;
        case 2: metrics_body<7>(in, mets, kidx, ch, b, hist, sums); break;
        case 3: metrics_body<9>(in, mets, kidx, ch, b, hist, sums); break;
        default: metrics_body<11>(in, mets, kidx, ch, b, hist, sums); break;
    }
}

// ---------------- Stage 2: bilinear resize as WMMA GEMM ----------------
// Out(224x224) = A(224x5) * M(5x5) * A^T(5x224). Per 16x16 tile:
// tB[p] = sum_q M[p][q] * wB(j,q) folds M*A^T; then D = A_rows * tB via
// two V_WMMA_F32_16X16X4_F32 k-steps (K padded 5 -> 8).

__global__ __launch_bounds__(32)
void fractal_resize_wmma_kernel(const float* __restrict__ mets, float* __restrict__ out) {
    const int tile = blockIdx.x;            // 0..195 (14x14 tiles)
    const int ch   = blockIdx.y;            // 0..2
    const int b    = blockIdx.z;            // 0..63
    const int ti   = tile / 14, tj = tile % 14;

    const int lane = threadIdx.x;           // wave32
    const int lr   = lane & 15;
    const int hi   = lane >> 4;             // 0: lanes 0-15, 1: lanes 16-31

    const int i = ti * 16 + lr;             // A-operand row (M = lane%16)
    const int j = tj * 16 + lr;             // B-operand col (N = lane%16)

    float wA[5], wB[5];
#pragma unroll
    for (int p = 0; p < 5; ++p) { wA[p] = interp_w(i, p); wB[p] = interp_w(j, p); }

    // tB[p] = (M * A^T)[p][j]; M loads are block-uniform -> scalar cached
    float tB[5];
#pragma unroll
    for (int p = 0; p < 5; ++p) {
        float acc = 0.0f;
#pragma unroll
        for (int q = 0; q < 5; ++q)
            acc += mets[(((size_t)b * 5 + p) * 5 + q) * NCH + ch] * wB[q];
        tB[p] = acc;
    }

    // 32-bit A 16x4 layout: VGPR0 = K0(lanes<16)/K2(lanes>=16), VGPR1 = K1/K3.
    // 32-bit B 4x16 layout mirrors it: lanes<16 hold K{0,1}, lanes>=16 hold K{2,3}.
    v2f a0, b0, a1, b1;
    a0.x = wA[2 * hi];     a0.y = wA[2 * hi + 1];
    b0.x = tB[2 * hi];     b0.y = tB[2 * hi + 1];
    a1.x = hi ? 0.0f : wA[4];  a1.y = 0.0f;     // k-step 1: K=4 valid, K=5..7 zero
    b1.x = hi ? 0.0f : tB[4];  b1.y = 0.0f;

    v8f c = {};
    c = __builtin_amdgcn_wmma_f32_16x16x4_f32(false, a0, false, b0, (short)0, c, false, false);
    c = __builtin_amdgcn_wmma_f32_16x16x4_f32(false, a1, false, b1, (short)0, c, false, false);

    // C/D layout: lane holds column N = lane%16; VGPR v holds row M = v + 8*hi
#pragma unroll
    for (int v = 0; v < 8; ++v) {
        int oi = ti * 16 + v + 8 * hi;
        out[(((size_t)b * IMG_H + oi) * IMG_W + j) * NCH + ch] = c[v];
    }
}

extern "C" void kernel_launch(void* const* d_in, const int* in_sizes, int n_in,
                              void* d_out, int out_size, void* d_ws, size_t ws_size,
                              hipStream_t stream) {
    const float* in  = (const float*)d_in[0];    // (64,224,224,3) f32
    float* out       = (float*)d_out;            // (64,224,224,3) f32
    float* mets      = (float*)d_ws;             // (64,5,5,3) f32 scratch

    dim3 g1(5, 3, IMG_B);
    fractal_metrics_kernel<<<g1, 256, 0, stream>>>(in, mets);

    dim3 g2(196, 3, IMG_B);
    fractal_resize_wmma_kernel<<<g2, 32, 0, stream>>>(mets, out);
}